// SplineNetwork_79731772883636
// MI455X (gfx1250) — compile-verified
//
#include <hip/hip_runtime.h>

// SplineNetwork forward on MI455X (gfx1250, wave32).
// Analytic uniform-grid reformulation of the reference's brute-force
// distance + top-16: the 16 nearest control points of the uniform 128x128
// grid are the 4x4 cell neighborhood; everything outside has cubic_conv==0.
// Weight table (64KB) staged in LDS (320KB/WGP). Final K=4 row reduction is
// done with V_WMMA_F32_16X16X4_F32 (B = all-ones), 16 points per wave.

#define GRID_N 128
#define NPTS   (GRID_N * GRID_N)   // 16384 weights / control points
#define NQ     16384               // query points
#define THREADS 256                // 8 waves
#define PTS_PER_BLOCK 128          // 8 waves * 16 points each

typedef __attribute__((ext_vector_type(2))) float v2f;
typedef __attribute__((ext_vector_type(8))) float v8f;

__device__ __forceinline__ float cubic_conv(float s) {
    float a  = fabsf(s);
    float f1 = (1.5f * a - 2.5f) * a * a + 1.0f;
    float f2 = ((-0.5f * a + 2.5f) * a - 4.0f) * a + 2.0f;
    float r  = (a < 1.0f) ? f1 : f2;
    // strict inequalities as in the reference: |s|==1 -> 0, |s|>=2 -> 0
    bool nz  = (a < 1.0f) | ((a > 1.0f) & (a < 2.0f));
    return nz ? r : 0.0f;
}

__global__ __launch_bounds__(THREADS)
void spline_fwd(const float* __restrict__ x,
                const float* __restrict__ w,
                float* __restrict__ out) {
    __shared__ float wlds[NPTS];   // 64 KB weight table in LDS

    const int t = threadIdx.x;

    // Stage weights into LDS (prefetch hint -> global_prefetch_b8)
    __builtin_prefetch(w + t, 0, 1);
#pragma unroll
    for (int i = 0; i < NPTS / THREADS; ++i)
        wlds[t + i * THREADS] = w[t + i * THREADS];
    __syncthreads();

    const int lane = t & 31;
    const int wv   = t >> 5;
    const int p16  = lane & 15;    // point within the wave's 16-point tile
    const int half = lane >> 4;    // 0: j-offsets {-1,0}, 1: j-offsets {+1,+2}

    const int ptBase = blockIdx.x * PTS_PER_BLOCK + wv * 16;
    const int pid    = ptBase + p16;

    const float qx = x[2 * pid + 0];
    const float qy = x[2 * pid + 1];

    const float inv_h = 63.5f;               // (N-1)/2
    const float u = (qx + 1.0f) * inv_h;     // grid-unit coords in [0,127]
    const float v = (qy + 1.0f) * inv_h;
    const int i1 = (int)floorf(u);
    const int j1 = (int)floorf(v);

    // 4 x-direction kernel values + clamped column indices
    float cx[4];
    int   ic[4];
#pragma unroll
    for (int o = 0; o < 4; ++o) {
        int ii   = i1 + o - 1;
        bool ok  = (ii >= 0) & (ii < GRID_N);
        cx[o]    = ok ? cubic_conv(u - (float)ii) : 0.0f;
        ic[o]    = min(max(ii, 0), GRID_N - 1);
    }

    // This lane's two row-partials -> A-matrix (16x4 f32 WMMA layout:
    // lanes 0-15 hold K=0,1 in VGPR0,1; lanes 16-31 hold K=2,3)
    v2f A;
#pragma unroll
    for (int r = 0; r < 2; ++r) {
        int  oj  = 2 * half - 1 + r;         // -1,0 or +1,+2
        int  jj  = j1 + oj;
        bool ok  = (jj >= 0) & (jj < GRID_N);
        float cy = ok ? cubic_conv(v - (float)jj) : 0.0f;
        int  jc  = min(max(jj, 0), GRID_N - 1);
        const float* row = &wlds[jc * GRID_N];
        float s = row[ic[0]] * cx[0] + row[ic[1]] * cx[1]
                + row[ic[2]] * cx[2] + row[ic[3]] * cx[3];
        A[r] = cy * s;
    }

    // D = A(16x4) x ones(4x16): every column of D equals the per-point sum.
    v2f Bones;
    Bones[0] = 1.0f;
    Bones[1] = 1.0f;
    v8f C = {};
    v8f D = __builtin_amdgcn_wmma_f32_16x16x4_f32(
        /*neg_a=*/false, A, /*neg_b=*/false, Bones,
        /*c_mod=*/(short)0, C, /*reuse_a=*/false, /*reuse_b=*/false);

    // D layout: lanes 0-15 VGPR r -> M=r ; lanes 16-31 VGPR r -> M=r+8.
    // Lane 0 holds out[0..7], lane 16 holds out[8..15] of this tile.
    if (p16 == 0) {
        float* o = out + ptBase + half * 8;
        *(v8f*)o = D;   // 32B-aligned (ptBase multiple of 16)
    }

    // Second tuple output: copy of x. 128 blocks * 256 threads == 32768 floats.
    const int g = blockIdx.x * THREADS + t;
    out[NQ + g] = x[g];
}

extern "C" void kernel_launch(void* const* d_in, const int* in_sizes, int n_in,
                              void* d_out, int out_size, void* d_ws, size_t ws_size,
                              hipStream_t stream) {
    const float* x = (const float*)d_in[0];       // [16384, 2]
    const float* w = (const float*)d_in[1];       // [16384, 1]
    // d_in[2] (control_points) is a uniform grid -> derived analytically.
    float* out = (float*)d_out;                   // [16384] output ++ [32768] x copy

    spline_fwd<<<NQ / PTS_PER_BLOCK, THREADS, 0, stream>>>(x, w, out);
}